// LightGCNMModel_65833258713793
// MI455X (gfx1250) — compile-verified
//
#include <hip/hip_runtime.h>
#include <math.h>

// ---------------------------------------------------------------------------
// LightGCN-style pipeline for MI455X (gfx1250, wave32).
//   proj GEMM (30000x896 @ 896x64) -> v_wmma_f32_16x16x4_f32 (exact f32)
//   3x graph propagation            -> L2-resident f32 atomic scatter
// ---------------------------------------------------------------------------

typedef float v2f __attribute__((ext_vector_type(2)));
typedef float v8f __attribute__((ext_vector_type(8)));

#define D0C 768
#define D1C 128
#define KDIM 896   // D0C + D1C
#define KN 64      // output embedding dim
#define KCH 128    // K-chunk staged through LDS

#if __has_builtin(__builtin_amdgcn_wmma_f32_16x16x4_f32)
#define HAVE_F32_WMMA 1
#else
#define HAVE_F32_WMMA 0
#endif

#if HAVE_F32_WMMA
// Each wave computes a 16x64 output tile: 4 x v_wmma_f32_16x16x4_f32 per K-step.
// A (16x4 f32): lanes 0-15 -> M=lane, K={k,k+1}; lanes 16-31 -> M=lane-16, K={k+2,k+3}.
// B (4x16 f32): lanes 0-15 -> N=lane, K={k,k+1}; lanes 16-31 -> N=lane-16, K={k+2,k+3}.
// C/D (16x16 f32): VGPR r -> M=r (lanes 0-15) / M=r+8 (lanes 16-31), N=lane%16.
__global__ __launch_bounds__(256) void lgcn_proj_wmma(
    const float* __restrict__ F0, const float* __restrict__ F1,
    const float* __restrict__ W, float* __restrict__ P, int I)
{
    __shared__ float sB[KCH * KN];  // transposed: sB[c*KCH + kl] = W[(c0+kl)*KN + c]
    const int tid    = threadIdx.x;
    const int wave   = tid >> 5;
    const int lane   = tid & 31;
    const int half   = lane >> 4;   // which K-pair this lane holds
    const int lanelo = lane & 15;

    const int r0     = blockIdx.x * 128 + wave * 16;
    const bool active = (r0 < I);                 // I % 16 == 0 -> whole tiles only
    const int  mrow   = active ? (r0 + lanelo) : lanelo;  // clamp inactive waves to row 0..15
    const float* pF0  = F0 + (size_t)mrow * D0C;
    const float* pF1  = F1 + (size_t)mrow * D1C;

    v8f acc0 = {}, acc1 = {}, acc2 = {}, acc3 = {};

    for (int c0 = 0; c0 < KDIM; c0 += KCH) {
        __syncthreads();
        // cooperative load of W[c0:c0+KCH][0:64], transposed so K is contiguous
        for (int idx = tid; idx < KCH * KN; idx += 256) {
            int kl = idx >> 6;   // 0..127
            int c  = idx & 63;
            sB[c * KCH + kl] = W[(size_t)(c0 + kl) * KN + c];
        }
        __syncthreads();
        const float* pA = (c0 < D0C) ? (pF0 + c0) : (pF1 + (c0 - D0C));
        for (int kk = 0; kk < KCH / 4; ++kk) {
            const int bofs = kk * 4 + 2 * half;
            v2f a  = *(const v2f*)(pA + bofs);
            v2f b0 = *(const v2f*)&sB[(0 * 16 + lanelo) * KCH + bofs];
            v2f b1 = *(const v2f*)&sB[(1 * 16 + lanelo) * KCH + bofs];
            v2f b2 = *(const v2f*)&sB[(2 * 16 + lanelo) * KCH + bofs];
            v2f b3 = *(const v2f*)&sB[(3 * 16 + lanelo) * KCH + bofs];
            acc0 = __builtin_amdgcn_wmma_f32_16x16x4_f32(false, a, false, b0, (short)0, acc0, false, false);
            acc1 = __builtin_amdgcn_wmma_f32_16x16x4_f32(false, a, false, b1, (short)0, acc1, false, false);
            acc2 = __builtin_amdgcn_wmma_f32_16x16x4_f32(false, a, false, b2, (short)0, acc2, false, false);
            acc3 = __builtin_amdgcn_wmma_f32_16x16x4_f32(false, a, false, b3, (short)0, acc3, false, false);
        }
    }
    if (!active) return;
#pragma unroll
    for (int r = 0; r < 8; ++r) {
        int row = r0 + r + 8 * half;
        float* pr = P + (size_t)row * KN + lanelo;
        pr[0]  = acc0[r];
        pr[16] = acc1[r];
        pr[32] = acc2[r];
        pr[48] = acc3[r];
    }
}
#else
// VALU fallback (only if the f32 WMMA builtin is not declared on this toolchain)
__global__ void lgcn_proj_wmma(const float* __restrict__ F0, const float* __restrict__ F1,
                               const float* __restrict__ W, float* __restrict__ P, int I)
{
    int idx = blockIdx.x * blockDim.x + threadIdx.x;
    if (idx >= I * KN) return;
    int row = idx >> 6, c = idx & 63;
    const float* f0 = F0 + (size_t)row * D0C;
    const float* f1 = F1 + (size_t)row * D1C;
    float s = 0.f;
    for (int k = 0; k < D0C; ++k) s = fmaf(f0[k], W[(size_t)k * KN + c], s);
    for (int k = 0; k < D1C; ++k) s = fmaf(f1[k], W[(size_t)(D0C + k) * KN + c], s);
    P[idx] = s;
}
#endif

// bias + row-wise L2 normalize; one wave per row (64 cols -> 2 per lane)
__global__ __launch_bounds__(256) void lgcn_norm_rows(
    const float* __restrict__ P, const float* __restrict__ bias,
    float* __restrict__ Xi, float* __restrict__ Ai, int I)
{
    int wave = threadIdx.x >> 5, lane = threadIdx.x & 31;
    int row = blockIdx.x * 8 + wave;
    if (row >= I) return;
    int c = lane * 2;
    float2 v = *(const float2*)&P[(size_t)row * KN + c];
    v.x += bias[c];
    v.y += bias[c + 1];
    float ssq = v.x * v.x + v.y * v.y;
    for (int m = 16; m >= 1; m >>= 1) ssq += __shfl_xor(ssq, m, 32);
    float n = fmaxf(sqrtf(ssq), 1e-12f);
    float inv = 1.0f / n;
    float2 o; o.x = v.x * inv; o.y = v.y * inv;
    *(float2*)&Xi[(size_t)row * KN + c] = o;
    *(float2*)&Ai[(size_t)row * KN + c] = o;
}

__global__ void lgcn_copy2(const float4* __restrict__ g, float4* __restrict__ a,
                           float4* __restrict__ b, int n4)
{
    int i = blockIdx.x * blockDim.x + threadIdx.x;
    if (i < n4) { float4 v = g[i]; a[i] = v; b[i] = v; }
}

__global__ void lgcn_zero4(float4* __restrict__ p, int n4)
{
    int i = blockIdx.x * blockDim.x + threadIdx.x;
    if (i < n4) p[i] = make_float4(0.f, 0.f, 0.f, 0.f);
}

__global__ void lgcn_add4(float4* __restrict__ a, const float4* __restrict__ b, int n4)
{
    int i = blockIdx.x * blockDim.x + threadIdx.x;
    if (i < n4) {
        float4 x = a[i]; float4 y = b[i];
        x.x += y.x; x.y += y.y; x.z += y.z; x.w += y.w;
        a[i] = x;
    }
}

__global__ void lgcn_deg(const int* __restrict__ dst, float* __restrict__ deg, int E)
{
    int e = blockIdx.x * blockDim.x + threadIdx.x;
    if (e < E) atomicAdd(&deg[dst[e]], 1.0f);
}

__global__ void lgcn_dinv(float* __restrict__ d, int N)
{
    int i = blockIdx.x * blockDim.x + threadIdx.x;
    if (i < N) {
        float v = d[i];
        d[i] = (v > 0.f) ? rsqrtf(fmaxf(v, 1.0f)) : 0.f;
    }
}

__global__ void lgcn_enorm(const int* __restrict__ src, const int* __restrict__ dst,
                           const float* __restrict__ dinv, float* __restrict__ w, int E)
{
    int e = blockIdx.x * blockDim.x + threadIdx.x;
    if (e < E) w[e] = dinv[src[e]] * dinv[dst[e]];
}

// one wave per edge: gather 64-f32 src row (L2-resident), scale, atomic scatter
__global__ __launch_bounds__(256) void lgcn_scatter(
    const int* __restrict__ src, const int* __restrict__ dst,
    const float* __restrict__ w, const float* __restrict__ cur,
    float* __restrict__ nxt, int E)
{
    int wave = threadIdx.x >> 5, lane = threadIdx.x & 31;
    int e = blockIdx.x * 8 + wave;
    if (e >= E) return;
    int s = src[e], d = dst[e];
    float wt = w[e];
    int c = lane * 2;
    float2 m = *(const float2*)&cur[(size_t)s * KN + c];
    atomicAdd(&nxt[(size_t)d * KN + c], m.x * wt);
    atomicAdd(&nxt[(size_t)d * KN + c + 1], m.y * wt);
}

// one wave per batch element; mean^2 of 4 layers folded into *1/16
__global__ __launch_bounds__(256) void lgcn_dot(
    const float* __restrict__ acc, const int* __restrict__ uidx,
    const int* __restrict__ iidx, float* __restrict__ out, int U, int B)
{
    int wave = threadIdx.x >> 5, lane = threadIdx.x & 31;
    int b = blockIdx.x * 8 + wave;
    if (b >= B) return;
    int u = uidx[b];
    int it = iidx[b];
    int c = lane * 2;
    float2 a = *(const float2*)&acc[(size_t)u * KN + c];
    float2 f = *(const float2*)&acc[(size_t)(U + it) * KN + c];
    float s = a.x * f.x + a.y * f.y;
    for (int m = 16; m >= 1; m >>= 1) s += __shfl_xor(s, m, 32);
    if (lane == 0) out[b] = s * 0.0625f;  // (1/4)*(1/4)
}

extern "C" void kernel_launch(void* const* d_in, const int* in_sizes, int n_in,
                              void* d_out, int out_size, void* d_ws, size_t ws_size,
                              hipStream_t stream)
{
    const float* Gu   = (const float*)d_in[0];
    const float* F0   = (const float*)d_in[1];
    const float* F1   = (const float*)d_in[2];
    const float* W    = (const float*)d_in[3];
    const float* bias = (const float*)d_in[4];
    const int* edge   = (const int*)d_in[5];
    const int* uidx   = (const int*)d_in[6];
    const int* iidx   = (const int*)d_in[7];
    float* out = (float*)d_out;

    const int U = in_sizes[0] / KN;     // 50000
    const int I = in_sizes[2] / D1C;    // 30000
    const int N = U + I;
    const int E = in_sizes[5] / 2;      // 2,000,000
    const int B = in_sizes[6];          // 4096
    const int NK = N * KN;

    float* acc  = (float*)d_ws;         // sum of 4 layer embeddings
    float* buf0 = acc + NK;
    float* buf1 = buf0 + NK;
    float* dinv = buf1 + NK;
    float* wq   = dinv + N;
    float* P    = buf1;                 // pre-normalize GEMM output aliases buf1

    const int* esrc = edge;
    const int* edst = edge + E;

    // 1) projection GEMM (WMMA)
#if HAVE_F32_WMMA
    lgcn_proj_wmma<<<(I + 127) / 128, 256, 0, stream>>>(F0, F1, W, P, I);
#else
    lgcn_proj_wmma<<<(I * KN + 255) / 256, 256, 0, stream>>>(F0, F1, W, P, I);
#endif

    // 2) x[:U] = Gu (also seed acc)
    {
        int n4 = U * KN / 4;
        lgcn_copy2<<<(n4 + 255) / 256, 256, 0, stream>>>(
            (const float4*)Gu, (float4*)buf0, (float4*)acc, n4);
    }

    // 3) x[U:] = l2norm(P + bias) (also seed acc)
    lgcn_norm_rows<<<(I + 7) / 8, 256, 0, stream>>>(
        P, bias, buf0 + (size_t)U * KN, acc + (size_t)U * KN, I);

    // 4) degrees -> dinv -> per-edge norms
    lgcn_zero4<<<(N / 4 + 255) / 256, 256, 0, stream>>>((float4*)dinv, N / 4);
    lgcn_deg<<<(E + 255) / 256, 256, 0, stream>>>(edst, dinv, E);
    lgcn_dinv<<<(N + 255) / 256, 256, 0, stream>>>(dinv, N);
    lgcn_enorm<<<(E + 255) / 256, 256, 0, stream>>>(esrc, edst, dinv, wq, E);

    // 5) 3 propagation layers (ping-pong buf0/buf1, accumulate into acc)
    float* cur = buf0;
    float* nxt = buf1;
    int n4 = NK / 4;
    for (int l = 0; l < 3; ++l) {
        lgcn_zero4<<<(n4 + 255) / 256, 256, 0, stream>>>((float4*)nxt, n4);
        lgcn_scatter<<<(E + 7) / 8, 256, 0, stream>>>(esrc, edst, wq, cur, nxt, E);
        lgcn_add4<<<(n4 + 255) / 256, 256, 0, stream>>>((float4*)acc, (const float4*)nxt, n4);
        float* t = cur; cur = nxt; nxt = t;
    }

    // 6) batched user·item dot products
    lgcn_dot<<<(B + 7) / 8, 256, 0, stream>>>(acc, uidx, iidx, out, U, B);
}